// Tiny_ConvCIFAR10_KAN_Generator_33921651704118
// MI455X (gfx1250) — compile-verified
//
#include <hip/hip_runtime.h>
#include <math.h>

typedef float v2f __attribute__((ext_vector_type(2)));
typedef float v8f __attribute__((ext_vector_type(8)));

// Extended uniform knot vector: g[i] = (i-2)*(2/3) - 1, i = 0..7  (matches reference GRID)
__host__ __device__ constexpr float knot(int i) {
    return (float)(i - 2) * (2.0f / 3.0f) - 1.0f;
}

// Quadratic (k=2) Cox-de Boor B-spline bases, 5 outputs, matching reference recursion.
__device__ __forceinline__ void bspl5(float x, float out[5]) {
    float b0[7];
#pragma unroll
    for (int i = 0; i < 7; ++i)
        b0[i] = (x >= knot(i) && x < knot(i + 1)) ? 1.0f : 0.0f;
    float b1[6];
#pragma unroll
    for (int i = 0; i < 6; ++i) {
        float l = (x - knot(i)) * (1.0f / (knot(i + 1) - knot(i)));
        float r = (knot(i + 2) - x) * (1.0f / (knot(i + 2) - knot(i + 1)));
        b1[i] = l * b0[i] + r * b0[i + 1];
    }
#pragma unroll
    for (int i = 0; i < 5; ++i) {
        float l = (x - knot(i)) * (1.0f / (knot(i + 2) - knot(i)));
        float r = (knot(i + 3) - x) * (1.0f / (knot(i + 3) - knot(i + 1)));
        out[i] = l * b1[i] + r * b1[i + 1];
    }
}

// Pack [base_w | spline_w] (zero-padded to 16 columns) into WMMA B-fragment order:
// wp[kk*64 + lane*2 + i] = Wcomb[kk*4 + (lane>>4)*2 + i][lane&15],  kk = 0..53.
// Run once per layer; conv waves then load B with one coalesced b64 per step.
__global__ void pack_weights_kernel(const float* __restrict__ base_w,
                                    const float* __restrict__ spline_w,
                                    float* __restrict__ wp, int Cout)
{
    const int idx = blockIdx.x * blockDim.x + threadIdx.x;   // 0 .. 3455
    if (idx >= 54 * 64) return;
    const int kk = idx >> 6, r = idx & 63;
    const int lane = r >> 1, i = r & 1;
    const int half = lane >> 4, n = lane & 15;
    const int k = kk * 4 + half * 2 + i;
    float wv = 0.0f;
    if (n < Cout)
        wv = (k < 36) ? base_w[n * 36 + k] : spline_w[n * 180 + (k - 36)];
    wp[idx] = wv;
}

// h = relu(x @ lin_w.T + lin_b) : [512,100] x [16,100] -> [512,16]
// One wave per 16-row tile. K = 100 = 25 steps of the 16x16x4 f32 WMMA.
__global__ __launch_bounds__(32) void lin_relu_kernel(
    const float* __restrict__ x, const float* __restrict__ w,
    const float* __restrict__ bias, float* __restrict__ out)
{
    const int lane = threadIdx.x;
    const int m = lane & 15;      // A row / B column / D column
    const int half = lane >> 4;   // K sub-offset select
    const int row = blockIdx.x * 16 + m;

    v8f acc = {};
#pragma unroll 5
    for (int kk = 0; kk < 25; ++kk) {
        const int kb = kk * 4 + half * 2;
        v2f a, b;
        a.x = x[row * 100 + kb];        // A[M=m][K=kb]
        a.y = x[row * 100 + kb + 1];    // A[M=m][K=kb+1]
        b.x = w[m * 100 + kb];          // B[K=kb][N=m]  (B = lin_w.T)
        b.y = w[m * 100 + kb + 1];
        acc = __builtin_amdgcn_wmma_f32_16x16x4_f32(
            false, a, false, b, (short)0, acc, false, false);
    }
    const float bn = bias[m];
#pragma unroll
    for (int r = 0; r < 8; ++r) {
        const int row2 = blockIdx.x * 16 + half * 8 + r;  // D rows: lanes0-15 M=0..7, lanes16-31 M=8..15
        out[row2 * 16 + m] = fmaxf(acc[r] + bn, 0.0f);
    }
}

// Fused up2 + im2col + [relu | bsplines] features + K=216 f32 WMMA GEMM (+ optional tanh).
// hin: [B, 4, W/2, W/2] (pre-upsample), hout: [B, Cout, W, W], W = 1<<lw.
// One wave handles a 4x4 spatial output tile of one image; the 16 outputs touch only a
// 4x4 input region x 4 channels = 64 pixels, whose features are computed once (stage A)
// and scattered into a location-major panel (stage B). B-matrix comes pre-packed from wp.
#define FSTRIDE 220   // padded row stride (dwords): 220 mod 64 = 28 -> conflict-free b64 reads

__global__ __launch_bounds__(32) void kan_conv_kernel(
    const float* __restrict__ hin, const float* __restrict__ wp,
    float* __restrict__ hout, int lw, int Cout, int do_tanh)
{
    __shared__ float sF[16 * FSTRIDE];   // feature panel, location-major: sF[m*FSTRIDE + k]
    __shared__ float sPix[64 * 6];       // per-pixel features: [c*16 + ry*4 + rx][relu, b0..b4]

    const int lane = threadIdx.x;
    const int m = lane & 15;             // tile-local output location (dy*4+dx)
    const int half = lane >> 4;
    const int W = 1 << lw;
    const int Win = W >> 1;
    const int lt = lw - 2;                           // log2(tiles per row), >= 0 for all layers
    const int T = blockIdx.x;
    const int x0 = (T & ((1 << lt) - 1)) << 2;
    const int y0 = ((T >> lt) & ((1 << lt) - 1)) << 2;
    const int bb = T >> (2 * lt);
    const int oy = (y0 - 1) >> 1;                    // input-region origin (may be -1)
    const int ox = (x0 - 1) >> 1;

    // ---- stage A: per-pixel features for the 4x4 input region x 4 channels (2 px/lane) ----
    // OOB region pixels get the features of p=0 == reference zero padding
    // (up-sampled tap OOB  <=>  (tap>>1) input index OOB).
#pragma unroll
    for (int t = 0; t < 2; ++t) {
        const int idx = lane + t * 32;               // 0..63
        const int c = idx >> 4, r = idx & 15;
        const int iy = oy + (r >> 2), ix = ox + (r & 3);
        float p = 0.0f;
        if (iy >= 0 && iy < Win && ix >= 0 && ix < Win)
            p = hin[(((size_t)bb * 4 + c) * Win + iy) * Win + ix];
        float bs[5];
        bspl5(p, bs);
        sPix[idx * 6 + 0] = fmaxf(p, 0.0f);
#pragma unroll
        for (int s = 0; s < 5; ++s) sPix[idx * 6 + 1 + s] = bs[s];
    }
    __syncthreads();

    // ---- stage B: scatter per-pixel features into the location-major 16x216 panel ----
    {
        const int dy = m >> 2, dx = m & 3;
        float* fRow = &sF[m * FSTRIDE];
#pragma unroll
        for (int j = 0; j < 18; ++j) {
            const int jj = half * 18 + j;            // patch index: c*9 + kh*3 + kw
            const int c = jj / 9, off = jj % 9;
            const int kh = off / 3, kw2 = off % 3;
            // region-relative input coords; independent of tile origin:
            const int ry = (dy + kh + 1) >> 1;
            const int rx = (dx + kw2 + 1) >> 1;
            const float* pp = &sPix[(c * 16 + ry * 4 + rx) * 6];
            fRow[jj] = pp[0];                        // relu feature
#pragma unroll
            for (int s = 0; s < 5; ++s)
                fRow[36 + jj * 5 + s] = pp[1 + s];   // spline features
        }
    }
    __syncthreads();

    // ---- 54-step f32 WMMA GEMM: D[16 loc x 16 ch] = F[16x216] x Wt[216x16] ----
    // A: one ds_load_b64 per step; B: one coalesced global_load_b64 (L0/L2-resident panel).
    v8f acc = {};
    const float* fRow = &sF[m * FSTRIDE];
    const float* wb = wp + lane * 2;
#pragma unroll 6
    for (int kk = 0; kk < 54; ++kk) {
        const int kb = kk * 4 + half * 2;
        v2f a = *(const v2f*)(fRow + kb);
        v2f b = *(const v2f*)(wb + kk * 64);
        acc = __builtin_amdgcn_wmma_f32_16x16x4_f32(
            false, a, false, b, (short)0, acc, false, false);
    }

    // ---- store (NCHW); lane holds output channel n = m, rows M = half*8 + r ----
    if (m < Cout) {
#pragma unroll
        for (int r = 0; r < 8; ++r) {
            const int M = half * 8 + r;
            const int yy = y0 + (M >> 2);
            const int xx = x0 + (M & 3);
            float v = acc[r];
            if (do_tanh) v = tanhf(v);
            hout[(((size_t)bb * Cout + m) * W + yy) * W + xx] = v;
        }
    }
}

extern "C" void kernel_launch(void* const* d_in, const int* in_sizes, int n_in,
                              void* d_out, int out_size, void* d_ws, size_t ws_size,
                              hipStream_t stream)
{
    const float* x    = (const float*)d_in[0];
    const float* linw = (const float*)d_in[1];
    const float* linb = (const float*)d_in[2];
    const float* bw1  = (const float*)d_in[3];
    const float* sw1  = (const float*)d_in[4];
    const float* bw2  = (const float*)d_in[5];
    const float* sw2  = (const float*)d_in[6];
    const float* bw3  = (const float*)d_in[7];
    const float* sw3  = (const float*)d_in[8];
    const float* bw4  = (const float*)d_in[9];
    const float* sw4  = (const float*)d_in[10];
    float* out = (float*)d_out;

    // workspace layout (f32): 4 packed weight panels (3456 each), then
    // h0 [512,16], h1 [512,4,4,4], h2 [512,4,8,8], h3 [512,4,16,16]
    float* wp1 = (float*)d_ws;
    float* wp2 = wp1 + 54 * 64;
    float* wp3 = wp2 + 54 * 64;
    float* wp4 = wp3 + 54 * 64;
    float* h0  = wp4 + 54 * 64;
    float* h1  = h0 + 512 * 16;
    float* h2  = h1 + 512 * 4 * 16;
    float* h3  = h2 + 512 * 4 * 64;

    pack_weights_kernel<<<27, 128, 0, stream>>>(bw1, sw1, wp1, 4);
    pack_weights_kernel<<<27, 128, 0, stream>>>(bw2, sw2, wp2, 4);
    pack_weights_kernel<<<27, 128, 0, stream>>>(bw3, sw3, wp3, 4);
    pack_weights_kernel<<<27, 128, 0, stream>>>(bw4, sw4, wp4, 3);

    lin_relu_kernel<<<32, 32, 0, stream>>>(x, linw, linb, h0);
    // layer 1: [512,4,2,2] -> up2 -> conv -> [512,4,4,4]   (512 tiles of 4x4)
    kan_conv_kernel<<<512,   32, 0, stream>>>(h0, wp1, h1, 2, 4, 0);
    // layer 2: -> [512,4,8,8]                              (2048 tiles)
    kan_conv_kernel<<<2048,  32, 0, stream>>>(h1, wp2, h2, 3, 4, 0);
    // layer 3: -> [512,4,16,16]                            (8192 tiles)
    kan_conv_kernel<<<8192,  32, 0, stream>>>(h2, wp3, h3, 4, 4, 0);
    // layer 4: -> [512,3,32,32] + tanh                     (32768 tiles)
    kan_conv_kernel<<<32768, 32, 0, stream>>>(h3, wp4, out, 5, 3, 1);
}